// InterDomainInfoNCELoss_79998060855554
// MI455X (gfx1250) — compile-verified
//
#include <hip/hip_runtime.h>
#include <hip/hip_bf16.h>

// InfoNCE (bidirectional) loss, N=8192, D=128, tau=0.1.
// sim_gf == sim_fg^T, so a single S = f@g^T GEMM feeds both directions:
//   loss = (1/N) * sum_i [ log(rowsum_i) + log(colsum_i) - 2*S_ii/tau ]
// Tiled fp32 WMMA (v_wmma_f32_16x16x4_f32) with exp applied in-register,
// row/col sums accumulated via global f32 atomics into d_ws.

#define NN      8192
#define DD      128
#define TAU_INV 10.0f
// exp(x/tau) = exp2(x * (1/tau) * log2(e))
#define EXP2_SCALE (TAU_INV * 1.44269504088896340736f)
#define LN2      0.69314718055994530942f
#define LDS_STRIDE 132   // 128 + 4 pad: fragment loads hit all 64 LDS banks

typedef float v2f __attribute__((ext_vector_type(2)));
typedef float v8f __attribute__((ext_vector_type(8)));

__global__ __launch_bounds__(256) void diag_init_kernel(
    const float* __restrict__ f, const float* __restrict__ g,
    float* __restrict__ rowsum, float* __restrict__ colsum,
    float* __restrict__ diag) {
  int i = blockIdx.x * blockDim.x + threadIdx.x;
  if (i >= NN) return;
  rowsum[i] = 0.0f;
  colsum[i] = 0.0f;
  const float4* fr = (const float4*)(f + (size_t)i * DD);
  const float4* gr = (const float4*)(g + (size_t)i * DD);
  float acc = 0.0f;
#pragma unroll
  for (int k = 0; k < DD / 4; ++k) {
    float4 a = fr[k];
    float4 b = gr[k];
    acc += a.x * b.x + a.y * b.y + a.z * b.z + a.w * b.w;
  }
  diag[i] = acc;
}

__global__ __launch_bounds__(256) void gemm_expsum_kernel(
    const float* __restrict__ f, const float* __restrict__ g,
    float* __restrict__ rowsum, float* __restrict__ colsum) {
  __shared__ float sA[128 * LDS_STRIDE];  // f rows  [m][k]
  __shared__ float sB[128 * LDS_STRIDE];  // g rows  [n][k]  (B = g^T tile)

  const int rowBase = blockIdx.y * 128;
  const int colBase = blockIdx.x * 128;
  const int tid = threadIdx.x;

  // Stage 128x128 f32 blocks of f and g into LDS (K=128 fits entirely).
  {
    const float4* gf = (const float4*)(f + (size_t)rowBase * DD);
    const float4* gg = (const float4*)(g + (size_t)colBase * DD);
#pragma unroll
    for (int it = 0; it < 16; ++it) {
      int idx = tid + it * 256;      // 0..4095 float4 slots (32 per row)
      int r = idx >> 5;
      int c4 = idx & 31;
      float4 va = gf[idx];
      float4 vb = gg[idx];
      *(float4*)&sA[r * LDS_STRIDE + c4 * 4] = va;
      *(float4*)&sB[r * LDS_STRIDE + c4 * 4] = vb;
    }
  }
  __syncthreads();

  const int lane = tid & 31;
  const int wave = tid >> 5;
  const int laneM = lane & 15;   // M (for A) / N (for B) index within tile
  const int laneHi = lane >> 4;  // K sub-offset selector (0 -> K0..1, 1 -> K2..3)
  const int wr = wave >> 1;      // 0..3 : 32-row strip of the block
  const int wc = wave & 1;       // 0..1 : 64-col strip of the block

  v8f acc[2][4];
#pragma unroll
  for (int i = 0; i < 2; ++i)
#pragma unroll
    for (int j = 0; j < 4; ++j)
#pragma unroll
      for (int r = 0; r < 8; ++r) acc[i][j][r] = 0.0f;

  const float* aP0 = &sA[(wr * 32 + 0 + laneM) * LDS_STRIDE + 2 * laneHi];
  const float* aP1 = &sA[(wr * 32 + 16 + laneM) * LDS_STRIDE + 2 * laneHi];
  const float* bP0 = &sB[(wc * 64 + 0 + laneM) * LDS_STRIDE + 2 * laneHi];
  const float* bP1 = &sB[(wc * 64 + 16 + laneM) * LDS_STRIDE + 2 * laneHi];
  const float* bP2 = &sB[(wc * 64 + 32 + laneM) * LDS_STRIDE + 2 * laneHi];
  const float* bP3 = &sB[(wc * 64 + 48 + laneM) * LDS_STRIDE + 2 * laneHi];

#pragma unroll
  for (int k = 0; k < 128; k += 4) {
    v2f a0 = *(const v2f*)(aP0 + k);
    v2f a1 = *(const v2f*)(aP1 + k);
    v2f b0 = *(const v2f*)(bP0 + k);
    v2f b1 = *(const v2f*)(bP1 + k);
    v2f b2 = *(const v2f*)(bP2 + k);
    v2f b3 = *(const v2f*)(bP3 + k);
    acc[0][0] = __builtin_amdgcn_wmma_f32_16x16x4_f32(false, a0, false, b0, (short)0, acc[0][0], false, false);
    acc[0][1] = __builtin_amdgcn_wmma_f32_16x16x4_f32(false, a0, false, b1, (short)0, acc[0][1], false, false);
    acc[0][2] = __builtin_amdgcn_wmma_f32_16x16x4_f32(false, a0, false, b2, (short)0, acc[0][2], false, false);
    acc[0][3] = __builtin_amdgcn_wmma_f32_16x16x4_f32(false, a0, false, b3, (short)0, acc[0][3], false, false);
    acc[1][0] = __builtin_amdgcn_wmma_f32_16x16x4_f32(false, a1, false, b0, (short)0, acc[1][0], false, false);
    acc[1][1] = __builtin_amdgcn_wmma_f32_16x16x4_f32(false, a1, false, b1, (short)0, acc[1][1], false, false);
    acc[1][2] = __builtin_amdgcn_wmma_f32_16x16x4_f32(false, a1, false, b2, (short)0, acc[1][2], false, false);
    acc[1][3] = __builtin_amdgcn_wmma_f32_16x16x4_f32(false, a1, false, b3, (short)0, acc[1][3], false, false);
  }

  // exp(S/tau) in-register: native v_exp_f32 (exp2), pre-scaled by log2(e)/tau
  v8f e[2][4];
#pragma unroll
  for (int i = 0; i < 2; ++i)
#pragma unroll
    for (int j = 0; j < 4; ++j)
#pragma unroll
      for (int r = 0; r < 8; ++r)
        e[i][j][r] = __builtin_amdgcn_exp2f(acc[i][j][r] * EXP2_SCALE);

  // Row sums: C layout -> VGPR r holds M=r (lanes 0-15) / M=r+8 (lanes 16-31),
  // N = lane&15. Reduce across N with xor-shuffles inside each 16-lane half.
#pragma unroll
  for (int tr = 0; tr < 2; ++tr) {
#pragma unroll
    for (int r = 0; r < 8; ++r) {
      float s = e[tr][0][r] + e[tr][1][r] + e[tr][2][r] + e[tr][3][r];
      s += __shfl_xor(s, 1, 32);
      s += __shfl_xor(s, 2, 32);
      s += __shfl_xor(s, 4, 32);
      s += __shfl_xor(s, 8, 32);
      if (laneM == 0) {
        int row = rowBase + wr * 32 + tr * 16 + r + 8 * laneHi;
        atomicAdd(&rowsum[row], s);
      }
    }
  }

  // Col sums: per-lane sum over r covers M=0..7 (lo half) / 8..15 (hi half);
  // xor-16 combines halves, lanes 0-15 then hold full column sums.
#pragma unroll
  for (int tc = 0; tc < 4; ++tc) {
    float s = 0.0f;
#pragma unroll
    for (int tr = 0; tr < 2; ++tr)
#pragma unroll
      for (int r = 0; r < 8; ++r) s += e[tr][tc][r];
    s += __shfl_xor(s, 16, 32);
    if (laneHi == 0) {
      int col = colBase + wc * 64 + tc * 16 + laneM;
      atomicAdd(&colsum[col], s);
    }
  }
}

__global__ __launch_bounds__(256) void finalize_kernel(
    const float* __restrict__ rowsum, const float* __restrict__ colsum,
    const float* __restrict__ diag, float* __restrict__ out) {
  __shared__ float red[256];
  float local = 0.0f;
  for (int i = threadIdx.x; i < NN; i += 256) {
    // v_log_f32 returns log2(x); natural log = ln2 * log2(x)
    local += LN2 * (__builtin_amdgcn_logf(rowsum[i]) +
                    __builtin_amdgcn_logf(colsum[i])) -
             2.0f * TAU_INV * diag[i];
  }
  red[threadIdx.x] = local;
  __syncthreads();
  for (int s = 128; s > 0; s >>= 1) {
    if ((int)threadIdx.x < s) red[threadIdx.x] += red[threadIdx.x + s];
    __syncthreads();
  }
  if (threadIdx.x == 0) out[0] = red[0] / (float)NN;
}

extern "C" void kernel_launch(void* const* d_in, const int* in_sizes, int n_in,
                              void* d_out, int out_size, void* d_ws, size_t ws_size,
                              hipStream_t stream) {
  const float* f = (const float*)d_in[0];
  const float* g = (const float*)d_in[1];
  float* rowsum = (float*)d_ws;          // [8192]
  float* colsum = rowsum + NN;           // [8192]
  float* diag = colsum + NN;             // [8192]

  diag_init_kernel<<<NN / 256, 256, 0, stream>>>(f, g, rowsum, colsum, diag);
  dim3 grid(NN / 128, NN / 128);         // 64 x 64 blocks of 128x128
  gemm_expsum_kernel<<<grid, 256, 0, stream>>>(f, g, rowsum, colsum);
  finalize_kernel<<<1, 256, 0, stream>>>(rowsum, colsum, diag, (float*)d_out);
}